// BasicConvolutionBlock_52175262712451
// MI455X (gfx1250) — compile-verified
//
#include <hip/hip_runtime.h>
#include <hip/hip_bf16.h>

typedef __attribute__((ext_vector_type(2))) float v2f;
typedef __attribute__((ext_vector_type(4))) float v4f;
typedef __attribute__((ext_vector_type(8))) float v8f;

#define CIN  32
#define COUT 64
#define KOFF 27
#define EPS  1e-5f

// ---------------------------------------------------------------------------
// Kernel 1: zero the output accumulator (d_out is poisoned by the harness).
// ---------------------------------------------------------------------------
__global__ void zero_out_kernel(float4* __restrict__ out, int n4) {
    int i = blockIdx.x * blockDim.x + threadIdx.x;
    if (i < n4) out[i] = make_float4(0.f, 0.f, 0.f, 0.f);
}

// ---------------------------------------------------------------------------
// Kernel 2: gather -> f32 WMMA GEMM -> atomic scatter-accumulate.
// grid.y = kernel offset k (0..26); each wave owns 16 kernel-map pairs.
//
// WMMA f32 16x16x4 layouts (wave32):
//   A (16x4): lanes 0-15 rows 0-15; VGPR0 = K=(lane>>4)*2, VGPR1 = K+1
//   B (4x16): lanes hold cols; VGPR0 = K=(lane>>4)*2 row, VGPR1 = K+1 row
//   C/D (16x16): VGPR j = row j + 8*(lane>>4), col = lane&15
// ---------------------------------------------------------------------------
__global__ void __launch_bounds__(256)
spconv_wmma_kernel(const float* __restrict__ feats,
                   const float* __restrict__ W,
                   const int*   __restrict__ in_idx,
                   const int*   __restrict__ out_idx,
                   float*       __restrict__ out,
                   int M_pairs)
{
    const int k    = blockIdx.y;
    const int lane = threadIdx.x & 31;
    const int wave = threadIdx.x >> 5;

    const int wavesPerBlock = blockDim.x >> 5;
    const int waveId        = blockIdx.x * wavesPerBlock + wave;
    const int waveStride    = gridDim.x * wavesPerBlock;

    const float* __restrict__ Wk = W      + (size_t)k * CIN * COUT;
    const int*   __restrict__ ik = in_idx  + (size_t)k * M_pairs;
    const int*   __restrict__ ok = out_idx + (size_t)k * M_pairs;

    const int row16 = lane & 15;          // matrix row (A) / col (B,C) slot
    const int kbase = (lane >> 4) << 1;   // 0 or 2: which K pair this half-wave holds

    // --- Preload all B fragments for this k: W[k] is 32x64, reused by every
    //     chunk this wave processes. 8 K-steps x 4 COUT tiles x v2f = 64 VGPRs.
    v2f b[8][4];
#pragma unroll
    for (int kk = 0; kk < 8; ++kk) {
#pragma unroll
        for (int t = 0; t < 4; ++t) {
            const int krow = kk * 4 + kbase;
            const int c    = t * 16 + row16;
            b[kk][t].x = Wk[(krow    ) * COUT + c];
            b[kk][t].y = Wk[(krow + 1) * COUT + c];
        }
    }

    const int nChunks = (M_pairs + 15) >> 4;

    for (int chunk = waveId; chunk < nChunks; chunk += waveStride) {
        const int base = chunk << 4;

        // Gather: this lane's A row (lanes 16-31 re-load same row, other K pair)
        const int mA   = min(base + row16, M_pairs - 1);
        const int gRow = ik[mA];
        const float* __restrict__ frow = feats + (size_t)gRow * CIN;

        v8f acc[4] = {v8f{}, v8f{}, v8f{}, v8f{}};

#pragma unroll
        for (int kk = 0; kk < 8; ++kk) {
            v2f a;
            a.x = frow[kk * 4 + kbase];
            a.y = frow[kk * 4 + kbase + 1];
#pragma unroll
            for (int t = 0; t < 4; ++t) {
                acc[t] = __builtin_amdgcn_wmma_f32_16x16x4_f32(
                    /*neg_a=*/false, a, /*neg_b=*/false, b[kk][t],
                    /*c_mod=*/(short)0, acc[t],
                    /*reuse_a=*/false, /*reuse_b=*/false);
            }
        }

        // Scatter-accumulate. Lane holds col = t*16 + (lane&15),
        // rows j + 8*(lane>>4). Collisions across pairs -> f32 atomics (L2).
        const int rsel = (lane >> 4) * 8;
#pragma unroll
        for (int j = 0; j < 8; ++j) {
            const int m = base + rsel + j;
            if (m < M_pairs) {
                const int orow = ok[m];
                float* dst = out + (size_t)orow * COUT + row16;
#pragma unroll
                for (int t = 0; t < 4; ++t) {
                    atomicAdd(dst + t * 16, acc[t][j]);
                }
            }
        }
    }
}

// ---------------------------------------------------------------------------
// Kernel 3: fused BatchNorm (inference) + ReLU, in place over d_out.
// ---------------------------------------------------------------------------
__global__ void bn_relu_kernel(float* __restrict__ out,
                               const float* __restrict__ gamma,
                               const float* __restrict__ beta,
                               const float* __restrict__ mean,
                               const float* __restrict__ var,
                               int total)
{
    int i = blockIdx.x * blockDim.x + threadIdx.x;
    if (i < total) {
        const int c   = i & (COUT - 1);
        const float s = __frsqrt_rn(var[c] + EPS) * gamma[c];
        const float v = (out[i] - mean[c]) * s + beta[c];
        out[i] = fmaxf(v, 0.0f);
    }
}

// ---------------------------------------------------------------------------
// Host launcher. Input order: feats, W, gamma, beta, run_mean, run_var,
//                             in_idx, out_idx
// ---------------------------------------------------------------------------
extern "C" void kernel_launch(void* const* d_in, const int* in_sizes, int n_in,
                              void* d_out, int out_size, void* d_ws, size_t ws_size,
                              hipStream_t stream)
{
    const float* feats    = (const float*)d_in[0];
    const float* W        = (const float*)d_in[1];
    const float* gamma    = (const float*)d_in[2];
    const float* beta     = (const float*)d_in[3];
    const float* run_mean = (const float*)d_in[4];
    const float* run_var  = (const float*)d_in[5];
    const int*   in_idx   = (const int*)d_in[6];
    const int*   out_idx  = (const int*)d_in[7];
    float*       out      = (float*)d_out;

    const int N = in_sizes[0] / CIN;        // 500000
    const int M = in_sizes[6] / KOFF;       // 250000
    const int total = N * COUT;             // == out_size

    // 1) zero accumulator (float4 vectorized)
    {
        const int n4 = total / 4;
        const int threads = 256;
        zero_out_kernel<<<(n4 + threads - 1) / threads, threads, 0, stream>>>(
            (float4*)out, n4);
    }

    // 2) gather + WMMA GEMM + atomic scatter
    {
        const int threads = 256;                 // 8 waves
        const int wavesPerBlock = threads / 32;
        const int nChunks = (M + 15) / 16;       // 15625
        const int gx = (nChunks + wavesPerBlock - 1) / wavesPerBlock;
        dim3 grid(gx, KOFF, 1);
        spconv_wmma_kernel<<<grid, threads, 0, stream>>>(
            feats, W, in_idx, out_idx, out, M);
    }

    // 3) BN + ReLU in place
    {
        const int threads = 256;
        bn_relu_kernel<<<(total + threads - 1) / threads, threads, 0, stream>>>(
            out, gamma, beta, run_mean, run_var, total);
    }
}